// DS1_37941741092924
// MI455X (gfx1250) — compile-verified
//
#include <hip/hip_runtime.h>

typedef __attribute__((ext_vector_type(2))) float v2f;
typedef __attribute__((ext_vector_type(8))) float v8f;

#define D_DIM 64
#define U_DIM 512
#define BT 128          // batch rows per block
#define UTILE 64        // units per block
#define LSTR 68         // LDS row stride in floats (bank-conflict-free, 16B aligned)

__global__ __launch_bounds__(256) void l2dist_wmma_kernel(
    const float* __restrict__ x, const float* __restrict__ w,
    float* __restrict__ out) {
  __shared__ float sx[BT * LSTR];      // 34816 B
  __shared__ float sw[UTILE * LSTR];   // 17408 B
  __shared__ float sxs[BT];
  __shared__ float sws[UTILE];

  const int tid = threadIdx.x;
  const int b0 = blockIdx.y * BT;
  const int u0 = blockIdx.x * UTILE;

  // ---- stage x tile (BT x 64 f32 = contiguous 32KB) into LDS via float4 ----
  {
    const float4* gx = (const float4*)(x + (size_t)b0 * D_DIM);
#pragma unroll
    for (int i = 0; i < (BT * D_DIM / 4) / 256; ++i) {
      int f = tid + i * 256;          // float4 index; 16 float4 per row
      int row = f >> 4;
      int c4 = f & 15;
      float4 v = gx[f];
      float* dst = &sx[row * LSTR + c4 * 4];
      dst[0] = v.x; dst[1] = v.y; dst[2] = v.z; dst[3] = v.w;
    }
  }
  // ---- stage w tile (UTILE x 64 f32 = contiguous 16KB) into LDS ----
  {
    const float4* gw = (const float4*)(w + (size_t)u0 * D_DIM);
#pragma unroll
    for (int i = 0; i < (UTILE * D_DIM / 4) / 256; ++i) {
      int f = tid + i * 256;
      int row = f >> 4;
      int c4 = f & 15;
      float4 v = gw[f];
      float* dst = &sw[row * LSTR + c4 * 4];
      dst[0] = v.x; dst[1] = v.y; dst[2] = v.z; dst[3] = v.w;
    }
  }
  __syncthreads();

  // ---- squared row norms from LDS ----
  if (tid < BT) {
    float s = 0.f;
#pragma unroll
    for (int k = 0; k < D_DIM; ++k) { float t = sx[tid * LSTR + k]; s += t * t; }
    sxs[tid] = s;
  } else if (tid < BT + UTILE) {
    int r = tid - BT;
    float s = 0.f;
#pragma unroll
    for (int k = 0; k < D_DIM; ++k) { float t = sw[r * LSTR + k]; s += t * t; }
    sws[r] = s;
  }
  __syncthreads();

  const int wave = tid >> 5;
  const int lane = tid & 31;
  const int mrow = lane & 15;               // M (for A) / N (for B)
  const int koff = (lane >> 4) << 1;        // K pair offset: 0 or 2
  const int rb = wave * 16;                 // this wave's batch-row base in tile

  // Preload all A fragments (K = 64 -> 16 k-steps, float2 per lane each).
  // A layout (16x4 f32): lane = M + 16*(K/2 half); VGPR0=K even, VGPR1=K odd.
  v2f a[16];
#pragma unroll
  for (int ks = 0; ks < 16; ++ks) {
    const float* p = &sx[(rb + mrow) * LSTR + ks * 4 + koff];
    a[ks].x = p[0];
    a[ks].y = p[1];
  }

#pragma unroll
  for (int ut = 0; ut < 4; ++ut) {
    const int ncol = ut * 16 + mrow;        // unit column within block tile
    v8f c = {};
#pragma unroll
    for (int ks = 0; ks < 16; ++ks) {
      // B layout (4x16 f32): lane = N + 16*(K/2 half); B[k][n] = w[u0+n][k]
      v2f b;
      const float* p = &sw[ncol * LSTR + ks * 4 + koff];
      b.x = p[0];
      b.y = p[1];
      c = __builtin_amdgcn_wmma_f32_16x16x4_f32(
          /*neg_a=*/false, a[ks], /*neg_b=*/false, b,
          /*c_mod=*/(short)0, c, /*reuse_a=*/false, /*reuse_b=*/false);
    }
    // Epilogue: out = ||x||^2 + ||w||^2 - 2*cross
    // C/D layout: VGPR v -> M = v + 8*(lane>=16), N = lane&15
    const float wsq = sws[ncol];
    const int h = (lane >> 4) << 3;         // 0 or 8
    float* orow = out + (size_t)(b0 + rb + h) * U_DIM + (u0 + ncol);
#pragma unroll
    for (int v = 0; v < 8; ++v) {
      float o = sxs[rb + h + v] + wsq - 2.0f * c[v];
      orow[(size_t)v * U_DIM] = o;
    }
  }
}

extern "C" void kernel_launch(void* const* d_in, const int* in_sizes, int n_in,
                              void* d_out, int out_size, void* d_ws, size_t ws_size,
                              hipStream_t stream) {
  const float* x = (const float*)d_in[0];   // [B, 64]
  const float* w = (const float*)d_in[1];   // [512, 64]
  float* out = (float*)d_out;               // [B, 512]
  const int batch = in_sizes[0] / D_DIM;    // 131072
  dim3 grid(U_DIM / UTILE, batch / BT);     // (8, 1024)
  l2dist_wmma_kernel<<<grid, 256, 0, stream>>>(x, w, out);
}